// LightGCN_87866440942255
// MI455X (gfx1250) — compile-verified
//
#include <hip/hip_runtime.h>
#include <hip/hip_bf16.h>
#include <stdint.h>

// ---------------------------------------------------------------------------
// LightGCN propagation for MI455X (gfx1250).
// Bandwidth/atomic-bound SpMM: CDNA5 async global->LDS staging of edge
// metadata (interleaved 16B records -> ds_load_b128 broadcast), float2
// gathers (L2-resident tables), native no-return f32 global atomics.
// ---------------------------------------------------------------------------

#define ALPHA 0.1f
#define D_FEAT 64
#define EPB 256   // edges per block
#define BLK 256   // threads per block (8 wave32)

// Generic pointer -> LDS byte offset (low 32 bits of flat LDS address).
typedef __attribute__((address_space(3))) void lds_void_t;
__device__ __forceinline__ unsigned lds_off(void* p) {
  return (unsigned)(unsigned long long)(lds_void_t*)p;
}

// CDNA5 async copy: 4 bytes global -> LDS per active lane (ASYNCcnt tracked).
__device__ __forceinline__ void async_load_b32(unsigned ldsoff, const void* gaddr) {
  asm volatile("global_load_async_to_lds_b32 %0, %1, off"
               :
               : "v"(ldsoff), "v"(gaddr)
               : "memory");
}

__device__ __forceinline__ void wait_asynccnt0() {
#if __has_builtin(__builtin_amdgcn_s_wait_asynccnt)
  __builtin_amdgcn_s_wait_asynccnt(0);
#else
  asm volatile("s_wait_asynccnt 0" ::: "memory");
#endif
}

// Relaxed agent-scope f32 atomic add -> global_atomic_add_f32 (no return).
__device__ __forceinline__ void atomic_add_f32(float* p, float v) {
  __hip_atomic_fetch_add(p, v, __ATOMIC_RELAXED, __HIP_MEMORY_SCOPE_AGENT);
}

// 16-byte interleaved edge record so one ds_load_b128 broadcast fetches all
// per-edge metadata.
struct __align__(16) EdgeMeta {
  int   u;
  int   i;
  float aui;
  float aiu;
};

// ---------------------------------------------------------------------------
// Init: L0 = emb0 ; L1 = L2 = ALPHA * emb0  (accumulator pre-seed, float4).
// ---------------------------------------------------------------------------
__global__ void lightgcn_init_kernel(const float4* __restrict__ emb0,
                                     float4* __restrict__ L0,
                                     float4* __restrict__ L1,
                                     float4* __restrict__ L2,
                                     int n4) {
  int t = blockIdx.x * blockDim.x + threadIdx.x;
  if (t < n4) {
    float4 v = emb0[t];
    L0[t] = v;
    float4 a = make_float4(ALPHA * v.x, ALPHA * v.y, ALPHA * v.z, ALPHA * v.w);
    L1[t] = a;
    L2[t] = a;
  }
}

// ---------------------------------------------------------------------------
// One propagation layer, both directions fused:
//   nextU[u] += a_ui[e] * prevI[i]   and   nextI[i] += a_iu[e] * prevU[u]
// Block stages 256 edges' metadata into LDS via async copies; each wave then
// processes 32 edges with lane-parallel float2 gathers + f32 atomics.
// ---------------------------------------------------------------------------
__global__ void __launch_bounds__(BLK)
lightgcn_scatter_kernel(const int*   __restrict__ edge_u,
                        const int*   __restrict__ edge_i,
                        const float* __restrict__ a_ui,
                        const float* __restrict__ a_iu,
                        const float* __restrict__ prevU,
                        const float* __restrict__ prevI,
                        float*       __restrict__ nextU,
                        float*       __restrict__ nextI,
                        int num_edges) {
  __shared__ EdgeMeta s_edge[EPB];   // 4 KB

  const int  tid = threadIdx.x;
  const long e0  = (long)blockIdx.x * EPB;
  const long ge  = e0 + tid;

  // Stage this block's edge metadata into LDS with async copies.
  // Global side stays fully coalesced per array; LDS side interleaves.
  if (ge < num_edges) {
    async_load_b32(lds_off(&s_edge[tid].u),   &edge_u[ge]);
    async_load_b32(lds_off(&s_edge[tid].i),   &edge_i[ge]);
    async_load_b32(lds_off(&s_edge[tid].aui), &a_ui[ge]);
    async_load_b32(lds_off(&s_edge[tid].aiu), &a_iu[ge]);
    // Stream the next tile toward L2 while this one chews on atomics.
    if (ge + EPB < num_edges) {
      __builtin_prefetch(&edge_u[ge + EPB], 0, 0);
      __builtin_prefetch(&edge_i[ge + EPB], 0, 0);
      __builtin_prefetch(&a_ui[ge + EPB], 0, 0);
      __builtin_prefetch(&a_iu[ge + EPB], 0, 0);
    }
  }
  wait_asynccnt0();
  __syncthreads();

  const int wave  = tid >> 5;       // 8 waves/block
  const int lane  = tid & 31;       // wave32
  const int d0    = lane * 2;       // each lane owns 2 of 64 feature dims
  const int ebase = wave * 32;      // 32 edges per wave

  const bool full_tile = (e0 + EPB) <= num_edges;

  if (full_tile) {
    // Unroll 4: batch ds_load_b128 metadata + b64 gathers ahead of the
    // 16 fire-and-forget atomics to maximize outstanding L2 traffic.
#pragma unroll 4
    for (int k = 0; k < 32; ++k) {
      const int4  m   = *(const int4*)&s_edge[ebase + k];   // broadcast
      const int   u   = m.x;
      const int   i   = m.y;
      const float aui = __int_as_float(m.z);
      const float aiu = __int_as_float(m.w);

      const float2 xi = *(const float2*)&prevI[(long)i * D_FEAT + d0];
      const float2 xu = *(const float2*)&prevU[(long)u * D_FEAT + d0];

      float* pu = &nextU[(long)u * D_FEAT + d0];
      float* pi = &nextI[(long)i * D_FEAT + d0];
      atomic_add_f32(pu + 0, aui * xi.x);
      atomic_add_f32(pu + 1, aui * xi.y);
      atomic_add_f32(pi + 0, aiu * xu.x);
      atomic_add_f32(pi + 1, aiu * xu.y);
    }
  } else {
#pragma unroll 1
    for (int k = 0; k < 32; ++k) {
      const int le = ebase + k;
      if (e0 + le >= num_edges) break;
      const int4  m   = *(const int4*)&s_edge[le];
      const int   u   = m.x;
      const int   i   = m.y;
      const float aui = __int_as_float(m.z);
      const float aiu = __int_as_float(m.w);

      const float2 xi = *(const float2*)&prevI[(long)i * D_FEAT + d0];
      const float2 xu = *(const float2*)&prevU[(long)u * D_FEAT + d0];

      float* pu = &nextU[(long)u * D_FEAT + d0];
      float* pi = &nextI[(long)i * D_FEAT + d0];
      atomic_add_f32(pu + 0, aui * xi.x);
      atomic_add_f32(pu + 1, aui * xi.y);
      atomic_add_f32(pi + 0, aiu * xu.x);
      atomic_add_f32(pi + 1, aiu * xu.y);
    }
  }
}

// ---------------------------------------------------------------------------
// kernel_launch
//   d_in: user_emb0, item_emb0, a_ui_vals, a_iu_vals, edge_u, edge_i
//   d_out: [3, N_USERS, 64] user stack then [3, N_ITEMS, 64] item stack (f32)
// ---------------------------------------------------------------------------
extern "C" void kernel_launch(void* const* d_in, const int* in_sizes, int n_in,
                              void* d_out, int out_size, void* d_ws, size_t ws_size,
                              hipStream_t stream) {
  const float* user_emb0 = (const float*)d_in[0];
  const float* item_emb0 = (const float*)d_in[1];
  const float* a_ui      = (const float*)d_in[2];
  const float* a_iu      = (const float*)d_in[3];
  const int*   edge_u    = (const int*)d_in[4];
  const int*   edge_i    = (const int*)d_in[5];

  const int n_users = in_sizes[0] / D_FEAT;
  const int n_items = in_sizes[1] / D_FEAT;
  const int E       = in_sizes[4];

  float* U0 = (float*)d_out;
  float* U1 = U0 + (size_t)n_users * D_FEAT;
  float* U2 = U1 + (size_t)n_users * D_FEAT;
  float* I0 = U2 + (size_t)n_users * D_FEAT;
  float* I1 = I0 + (size_t)n_items * D_FEAT;
  float* I2 = I1 + (size_t)n_items * D_FEAT;

  // Init layer-0 copies and alpha-seeded accumulators.
  const int nu4 = n_users * D_FEAT / 4;
  const int ni4 = n_items * D_FEAT / 4;
  lightgcn_init_kernel<<<(nu4 + BLK - 1) / BLK, BLK, 0, stream>>>(
      (const float4*)user_emb0, (float4*)U0, (float4*)U1, (float4*)U2, nu4);
  lightgcn_init_kernel<<<(ni4 + BLK - 1) / BLK, BLK, 0, stream>>>(
      (const float4*)item_emb0, (float4*)I0, (float4*)I1, (float4*)I2, ni4);

  const int nblk = (E + EPB - 1) / EPB;
  // Layer 1: reads L0, accumulates into L1.
  lightgcn_scatter_kernel<<<nblk, BLK, 0, stream>>>(
      edge_u, edge_i, a_ui, a_iu, U0, I0, U1, I1, E);
  // Layer 2: reads L1, accumulates into L2.
  lightgcn_scatter_kernel<<<nblk, BLK, 0, stream>>>(
      edge_u, edge_i, a_ui, a_iu, U1, I1, U2, I2, E);
}